// VectorQuantizer_78821239816220
// MI455X (gfx1250) — compile-verified
//
#include <hip/hip_runtime.h>
#include <hip/hip_bf16.h>

typedef __attribute__((ext_vector_type(2))) float v2f;
typedef __attribute__((ext_vector_type(8))) float v8f;
typedef __attribute__((ext_vector_type(4))) int   v4i;

// address-space-qualified views for the async global->LDS builtin
typedef __attribute__((address_space(1))) v4i gv4i;   // global
typedef __attribute__((address_space(3))) v4i lv4i;   // LDS

#define VQ_N      32768
#define VQ_D      256
#define VQ_K      1024
#define TILE_N    16
#define NTILES    (VQ_K / TILE_N)   // 64
#define KSTEPS    (VQ_D / 4)        // 64

// ---- CDNA5 async global->LDS path (gfx1250), with portable fallback --------
#if defined(__has_builtin)
#if __has_builtin(__builtin_amdgcn_global_load_async_to_lds_b128) && \
    __has_builtin(__builtin_amdgcn_s_wait_asynccnt)
#define VQ_HAVE_ASYNC_LDS 1
#endif
#endif
#ifndef VQ_HAVE_ASYNC_LDS
#define VQ_HAVE_ASYNC_LDS 0
#endif

// Copy one 16KB codebook tile (16 codes x 256 f32, contiguous) into LDS.
// 256 threads x 4 x 16B. Async variant never touches VGPR data registers.
__device__ __forceinline__ void vq_copy_tile(const float* __restrict__ gsrc,
                                             float4* __restrict__ ldst) {
#if VQ_HAVE_ASYNC_LDS
  gv4i* g = (gv4i*)gsrc + threadIdx.x;   // const cast + AS1 view
  lv4i* l = (lv4i*)ldst + threadIdx.x;   // AS3 view of __shared__
  #pragma unroll
  for (int i = 0; i < 4; ++i)
    __builtin_amdgcn_global_load_async_to_lds_b128(g + 256 * i, l + 256 * i, 0, 0);
#else
  const float4* src = (const float4*)gsrc;
  #pragma unroll
  for (int i = 0; i < 4; ++i)
    ldst[threadIdx.x + 256 * i] = src[threadIdx.x + 256 * i];
#endif
}

__device__ __forceinline__ void vq_copy_fence() {
#if VQ_HAVE_ASYNC_LDS
  __builtin_amdgcn_s_wait_asynccnt(0);
#endif
}

// ---------------------------------------------------------------------------
// Kernel 1: codebook row norms  ||c_k||^2  -> cnorm[1024]
// ---------------------------------------------------------------------------
__global__ __launch_bounds__(256) void vq_cnorm_kernel(
    const float* __restrict__ cb, float* __restrict__ cnorm) {
  const int lane = threadIdx.x & 31;
  const int wave = threadIdx.x >> 5;
  const int k = blockIdx.x * 8 + wave;               // 128 blocks * 8 waves
  const float* row = cb + (size_t)k * VQ_D;
  float s = 0.0f;
  #pragma unroll
  for (int i = 0; i < VQ_D / 32; ++i) {
    float v = row[lane + 32 * i];
    s += v * v;
  }
  #pragma unroll
  for (int m = 16; m >= 1; m >>= 1) s += __shfl_xor(s, m, 32);
  if (lane == 0) cnorm[k] = s;
}

// ---------------------------------------------------------------------------
// Kernel 2: WMMA distance + argmin.
//   Each block: 8 waves x 16 rows = 128 rows. 256 blocks cover N=32768.
//   A (16x256 fp32 per wave) held in VGPRs in the 16x4 fp32 WMMA layout.
//   B (16 codes x 256) staged in LDS via async DMA, double buffered.
//   dist candidate = ||c||^2 - 2*(x.c)   (||x||^2 constant per row -> dropped)
// ---------------------------------------------------------------------------
__global__ __launch_bounds__(256) void vq_argmin_kernel(
    const float* __restrict__ x, const float* __restrict__ cb,
    const float* __restrict__ cnorm, float* __restrict__ idx_out_f) {
  __shared__ float4 btile[2][TILE_N * VQ_D / 4];     // 2 x 16KB

  const int lane = threadIdx.x & 31;
  const int wave = threadIdx.x >> 5;                 // 0..7
  const int half = lane >> 4;                        // 0,1
  const int l15  = lane & 15;
  const long rowBase = (long)blockIdx.x * 128 + wave * 16;

  // kick off tile 0 DMA before spending time on A-fragment loads
  vq_copy_tile(cb, btile[0]);

  // ---- load A fragments: lane l15 holds row (rowBase+l15);
  //      lanes 0-15 hold dims {4s,4s+1}, lanes 16-31 hold {4s+2,4s+3}
  const float* xrow = x + (rowBase + l15) * VQ_D + 2 * half;
  v2f a[KSTEPS];
  #pragma unroll
  for (int s = 0; s < KSTEPS; ++s)
    a[s] = *(const v2f*)(xrow + 4 * s);

  float    minv[8];
  unsigned mini[8];
  #pragma unroll
  for (int j = 0; j < 8; ++j) { minv[j] = 3.4e38f; mini[j] = 0u; }

  int buf = 0;
  for (int t = 0; t < NTILES; ++t) {
    vq_copy_fence();   // our async stores to LDS complete
    __syncthreads();   // tile t visible to all waves; buf^1 reads retired

    // DMA next codebook tile into the other buffer while we compute
    if (t + 1 < NTILES)
      vq_copy_tile(cb + (size_t)(t + 1) * TILE_N * VQ_D, btile[buf ^ 1]);

    // ---- 16x16 dot tile: 64 fp32 WMMAs over D=256, two accumulator chains
    const float* bp = ((const float*)btile[buf]) + l15 * VQ_D + 2 * half;
    v8f acc0 = {0.f, 0.f, 0.f, 0.f, 0.f, 0.f, 0.f, 0.f};
    v8f acc1 = {0.f, 0.f, 0.f, 0.f, 0.f, 0.f, 0.f, 0.f};
    #pragma unroll
    for (int s = 0; s < KSTEPS; s += 2) {
      v2f b0 = *(const v2f*)(bp + 4 * s);
      v2f b1 = *(const v2f*)(bp + 4 * (s + 1));
      acc0 = __builtin_amdgcn_wmma_f32_16x16x4_f32(
          false, a[s], false, b0, (short)0, acc0, false, false);
      acc1 = __builtin_amdgcn_wmma_f32_16x16x4_f32(
          false, a[s + 1], false, b1, (short)0, acc1, false, false);
    }

    // ---- fold into running argmin (N = l15 for every acc element)
    const unsigned code = (unsigned)(t * TILE_N + l15);
    const float cn = cnorm[code];
    #pragma unroll
    for (int j = 0; j < 8; ++j) {
      float dval = cn - 2.0f * (acc0[j] + acc1[j]);
      if (dval < minv[j]) { minv[j] = dval; mini[j] = code; }
    }

    buf ^= 1;
  }

  // ---- cross-lane min over the 16 lanes holding one row's N values.
  //      acc layout: VGPR j, half h -> row M = j + 8*h, col N = l15.
  //      Tie-break: lower code index wins (matches jnp.argmin).
  #pragma unroll
  for (int j = 0; j < 8; ++j) {
    float v = minv[j];
    unsigned i = mini[j];
    #pragma unroll
    for (int m = 1; m <= 8; m <<= 1) {
      float    ov = __shfl_xor(v, m, 16);
      unsigned oi = __shfl_xor(i, m, 16);
      if (ov < v || (ov == v && oi < i)) { v = ov; i = oi; }
    }
    if (l15 == 0)
      idx_out_f[rowBase + j + 8 * half] = (float)i;
  }
}

// ---------------------------------------------------------------------------
// Kernel 3: gather quantized rows + per-block sum of (q - x)^2
// ---------------------------------------------------------------------------
__global__ __launch_bounds__(256) void vq_gather_kernel(
    const float* __restrict__ x, const float* __restrict__ cb,
    const float* __restrict__ idx_f, float* __restrict__ q_out,
    float* __restrict__ partials) {
  const int lane = threadIdx.x & 31;
  const int wave = threadIdx.x >> 5;
  const long row = (long)blockIdx.x * 8 + wave;      // 4096 blocks
  const int code = (int)idx_f[row];

  const float4* src = (const float4*)(cb + (size_t)code * VQ_D);
  const float4* xs  = (const float4*)(x + row * VQ_D);
  float4*       dst = (float4*)(q_out + row * VQ_D);

  float ss = 0.0f;
  #pragma unroll
  for (int i = 0; i < 2; ++i) {                      // 64 float4 per row / 32 lanes
    float4 q  = src[lane + 32 * i];
    float4 xv = xs[lane + 32 * i];
    dst[lane + 32 * i] = q;
    float dx = q.x - xv.x, dy = q.y - xv.y, dz = q.z - xv.z, dw = q.w - xv.w;
    ss += dx * dx + dy * dy + dz * dz + dw * dw;
  }

  #pragma unroll
  for (int m = 16; m >= 1; m >>= 1) ss += __shfl_xor(ss, m, 32);
  __shared__ float red[8];
  if (lane == 0) red[wave] = ss;
  __syncthreads();
  if (threadIdx.x == 0) {
    float t = 0.0f;
    #pragma unroll
    for (int w = 0; w < 8; ++w) t += red[w];
    partials[blockIdx.x] = t;
  }
}

// ---------------------------------------------------------------------------
// Kernel 4: deterministic final loss reduction.
//   loss = q_latent + 0.25*e_latent = 1.25 * mean((q - x)^2)
// ---------------------------------------------------------------------------
__global__ __launch_bounds__(256) void vq_loss_kernel(
    const float* __restrict__ partials, int n, float* __restrict__ loss) {
  __shared__ float red[256];
  float s = 0.0f;
  for (int i = threadIdx.x; i < n; i += 256) s += partials[i];
  red[threadIdx.x] = s;
  __syncthreads();
  for (int off = 128; off > 0; off >>= 1) {
    if (threadIdx.x < off) red[threadIdx.x] += red[threadIdx.x + off];
    __syncthreads();
  }
  if (threadIdx.x == 0)
    *loss = 1.25f * red[0] / (float)(VQ_N * (double)VQ_D);
}

// ---------------------------------------------------------------------------
extern "C" void kernel_launch(void* const* d_in, const int* in_sizes, int n_in,
                              void* d_out, int out_size, void* d_ws, size_t ws_size,
                              hipStream_t stream) {
  (void)in_sizes; (void)n_in; (void)out_size; (void)ws_size;
  const float* x  = (const float*)d_in[0];   // [32,32,32,256] f32
  const float* cb = (const float*)d_in[1];   // [1024,256]     f32

  // d_out layout: quantized_st [32768*256] | loss [1] | indices [32768] (as f32)
  float* q_out  = (float*)d_out;
  float* loss   = (float*)d_out + (size_t)VQ_N * VQ_D;
  float* idx_f  = (float*)d_out + (size_t)VQ_N * VQ_D + 1;

  float* cnorm    = (float*)d_ws;            // 1024 floats
  float* partials = (float*)d_ws + VQ_K;     // 4096 floats

  vq_cnorm_kernel <<<VQ_K / 8, 256, 0, stream>>>(cb, cnorm);
  vq_argmin_kernel<<<VQ_N / 128, 256, 0, stream>>>(x, cb, cnorm, idx_f);
  vq_gather_kernel<<<VQ_N / 8, 256, 0, stream>>>(x, cb, idx_f, q_out, partials);
  vq_loss_kernel  <<<1, 256, 0, stream>>>(partials, VQ_N / 8, loss);
}